// EmbedGCN_45286135169458
// MI455X (gfx1250) — compile-verified
//
#include <hip/hip_runtime.h>
#include <math.h>

typedef __attribute__((ext_vector_type(2))) float v2f;
typedef __attribute__((ext_vector_type(8))) float v8f;

#define B_   8
#define N_   500
#define K_   16
#define D_   32
#define C_   16
#define TIN  12
#define TOUT 12

// Hardware tanh on CDNA5 (V_TANH_F32, a TRANS32 op that co-executes with WMMA);
// fall back to libm if the builtin is unavailable in this toolchain.
__device__ __forceinline__ float fast_tanh(float x) {
#if __has_builtin(__builtin_amdgcn_tanhf)
    return __builtin_amdgcn_tanhf(x);
#elif __has_builtin(__builtin_amdgcn_tanh_f32)
    return __builtin_amdgcn_tanh_f32(x);
#else
    return tanhf(x);
#endif
}

// ---------------------------------------------------------------------------
// Kernel 1: k = feat @ Wk + bk ; q = feat @ Wq + bq   (tiny: 4000x32x32)
// one thread per (row, d) output element, computes both k and q.
// ---------------------------------------------------------------------------
__global__ void proj_kq_kernel(const float* __restrict__ feat,
                               const float* __restrict__ Wk, const float* __restrict__ bk,
                               const float* __restrict__ Wq, const float* __restrict__ bq,
                               float* __restrict__ kv, float* __restrict__ qv) {
    int gid = blockIdx.x * blockDim.x + threadIdx.x;
    if (gid >= B_ * N_ * D_) return;
    int row = gid >> 5;     // D_ == 32
    int d   = gid & 31;
    const float* f = feat + row * D_;
    float sk = bk[d];
    float sq = bq[d];
#pragma unroll
    for (int dd = 0; dd < D_; ++dd) {
        float fv = f[dd];
        sk += fv * Wk[dd * D_ + d];
        sq += fv * Wq[dd * D_ + d];
    }
    kv[gid] = sk;
    qv[gid] = sq;
}

// ---------------------------------------------------------------------------
// Kernel 2: one wave (32 lanes) per node (b,n).
//  Phase A: logits for the node's 16 neighbors via WMMA f32 16x16x4:
//           A (16 pairs x 32) = tanh(k[b,n,:] + q[b,j_m,:]), B = Wc (32x16)
//  Phase B: argmax over classes (cross-lane shfl_xor reduction, 16-lane halves)
//  Phase C: per-class masked mean over neighbors + per-class 12x12 linear+tanh
// ---------------------------------------------------------------------------
__global__ void __launch_bounds__(256)
node_kernel(const float* __restrict__ kv, const float* __restrict__ qv,
            const float* __restrict__ Wc, const float* __restrict__ bc,
            const float* __restrict__ input_seq, const float* __restrict__ Wcls,
            const int* __restrict__ adj, float* __restrict__ out) {
    const int wave = threadIdx.x >> 5;           // 8 waves / block
    const int lane = threadIdx.x & 31;
    const int node = blockIdx.x * 8 + wave;      // grid sized exactly: B_*N_ nodes
    if (node >= B_ * N_) return;
    const int b = node / N_;

    __shared__ int   s_j [8][K_];
    __shared__ int   s_id[8][K_];
    __shared__ float s_x [8][K_][TIN];

    const int half = lane >> 4;                  // 0: K-dims {0,1}; 1: K-dims {2,3} per step
    const int m    = lane & 15;                  // A row (neighbor) / B,C column (class)

    if (lane < 16) s_j[wave][lane] = adj[node * K_ + lane];
    __syncthreads();

    // cooperative load of x_topk (16 neighbors x 12 timesteps) into LDS
    for (int i = lane; i < K_ * TIN; i += 32) {
        int nb = i / TIN, t = i % TIN;
        s_x[wave][nb][t] = input_seq[(b * N_ + s_j[wave][nb]) * TIN + t];
    }

    // ---- Phase A: WMMA logits ---------------------------------------------
    const int    j    = s_j[wave][m];            // neighbor of row m
    const float* krow = kv + node * D_;          // k[b, n, :]
    const float* qrow = qv + (b * N_ + j) * D_;  // q[b, j, :]

    v8f acc = {};
#pragma unroll
    for (int kk = 0; kk < 8; ++kk) {
        const int c0 = kk * 4 + half * 2;        // this lane's two K-dims this step
        v2f kp = *(const v2f*)(krow + c0);       // 8B-aligned pair -> global_load_b64
        v2f qp = *(const v2f*)(qrow + c0);
        v2f a, w;
        a.x = fast_tanh(kp.x + qp.x);
        a.y = fast_tanh(kp.y + qp.y);
        w.x = Wc[(c0)     * C_ + m];             // B[k][class]
        w.y = Wc[(c0 + 1) * C_ + m];
        acc = __builtin_amdgcn_wmma_f32_16x16x4_f32(
            /*neg_a=*/false, a, /*neg_b=*/false, w,
            /*c_mod=*/(short)0, acc, /*reuse_a=*/false, /*reuse_b=*/false);
    }

    // ---- Phase B: argmax over classes -------------------------------------
    // lane holds class column N = m for rows M = r + 8*half; add bias per lane.
    const float bcv = bc[m];
    int myid[8];
#pragma unroll
    for (int r = 0; r < 8; ++r) {
        float v   = acc[r] + bcv;
        int   idx = m;
#pragma unroll
        for (int mask = 1; mask <= 8; mask <<= 1) {   // stays within 16-lane half
            float ov = __shfl_xor(v, mask, 32);
            int   oi = __shfl_xor(idx, mask, 32);
            if (ov > v || (ov == v && oi < idx)) { v = ov; idx = oi; }
        }
        myid[r] = idx;                                // first-max tie-break, like jnp.argmax
    }
    if (m == 0) {
#pragma unroll
        for (int r = 0; r < 8; ++r) s_id[wave][half * 8 + r] = myid[r];
    }
    __syncthreads();

    // ---- Phase C: masked mean + per-class Linear + tanh -------------------
    // lane -> class c = m; halves split the 12 output timesteps (6 each).
    const int c = m;
    float cnt = 0.0f;
    float sum[TIN];
#pragma unroll
    for (int t = 0; t < TIN; ++t) sum[t] = 0.0f;
    for (int nb = 0; nb < K_; ++nb) {
        if (s_id[wave][nb] == c) {
            cnt += 1.0f;
#pragma unroll
            for (int t = 0; t < TIN; ++t) sum[t] += s_x[wave][nb][t];
        }
    }
    const float inv = (cnt == 0.0f) ? 1.0f : (1.0f / cnt);

    float* orow = out + (node * C_ + c) * TOUT;
#pragma unroll
    for (int u = 0; u < 6; ++u) {
        const int to = half * 6 + u;
        float s = 0.0f;
#pragma unroll
        for (int t = 0; t < TIN; ++t)
            s += (sum[t] * inv) * Wcls[(c * TIN + t) * TOUT + to];
        orow[to] = fast_tanh(s);
    }
}

// ---------------------------------------------------------------------------
extern "C" void kernel_launch(void* const* d_in, const int* in_sizes, int n_in,
                              void* d_out, int out_size, void* d_ws, size_t ws_size,
                              hipStream_t stream) {
    const float* feat      = (const float*)d_in[0];  // (B,N,D)
    const float* input_seq = (const float*)d_in[1];  // (B,N,T_IN)
    const float* Wk        = (const float*)d_in[2];  // (D,D)
    const float* bk        = (const float*)d_in[3];  // (D,)
    const float* Wq        = (const float*)d_in[4];  // (D,D)
    const float* bq        = (const float*)d_in[5];  // (D,)
    const float* Wc        = (const float*)d_in[6];  // (D,C)
    const float* bc        = (const float*)d_in[7];  // (C,)
    const float* Wcls      = (const float*)d_in[8];  // (C,T_IN,T_OUT)
    const int*   adj       = (const int*)d_in[9];    // (B,N,K)
    float*       out       = (float*)d_out;          // (B,N,C,T_OUT)

    float* kv = (float*)d_ws;                 // B*N*D floats
    float* qv = kv + (size_t)B_ * N_ * D_;    // B*N*D floats (total 1 MB of ws)

    // Kernel 1: projections (B*N*D = 128000 threads)
    {
        const int total = B_ * N_ * D_;
        const int blk = 256;
        proj_kq_kernel<<<(total + blk - 1) / blk, blk, 0, stream>>>(
            feat, Wk, bk, Wq, bq, kv, qv);
    }
    // Kernel 2: one wave per node, 8 waves per block (B*N = 4000 nodes exactly)
    {
        const int blocks = (B_ * N_) / 8;     // 500
        node_kernel<<<blocks, 256, 0, stream>>>(
            kv, qv, Wc, bc, input_seq, Wcls, adj, out);
    }
}